// RRGraphConv_72344429133898
// MI455X (gfx1250) — compile-verified
//
#include <hip/hip_runtime.h>
#include <hip/hip_bf16.h>

#define N_NODES 100000
#define N_EDGES 1600000
#define D_IN    64
#define D_OUT   128
#define M_TILES_PER_WAVE 5          // 6250 m-tiles / 5 = 1250 chunks (exact)

typedef __attribute__((ext_vector_type(2))) float v2f;
typedef __attribute__((ext_vector_type(8))) float v8f;

// ---------------------------------------------------------------------------
// Kernel A: h = feat  (initializes the (1+EPS)*feat term, EPS==0)
// ---------------------------------------------------------------------------
__global__ void rrgc_init_h(const float* __restrict__ feat, float* __restrict__ h) {
    int t = blockIdx.x * blockDim.x + threadIdx.x;          // float4 granularity
    const int n4 = N_NODES * D_IN / 4;                      // 1.6M
    if (t < n4) {
        ((float4*)h)[t] = ((const float4*)feat)[t];
    }
}

// ---------------------------------------------------------------------------
// Kernel B: h[dst[e]] += feat[src[e]]  (mailbox scatter-add, weight == 1)
// One thread handles one edge x 2 channels (float2 gather + 2 f32 atomics).
// feat (25.6MB) + h (25.6MB) both fit in the 192MB L2, so these are
// L2-resident non-returning atomics (STOREcnt path, no VGPR writeback).
// ---------------------------------------------------------------------------
__global__ void rrgc_scatter(const float* __restrict__ feat,
                             const int*   __restrict__ src,
                             const int*   __restrict__ dst,
                             float*       __restrict__ h) {
    long long t = (long long)blockIdx.x * blockDim.x + threadIdx.x;
    const long long total = (long long)N_EDGES * (D_IN / 2);     // 51.2M
    if (t >= total) return;
    int e = (int)(t >> 5);                 // D_IN/2 == 32 lanes per edge
    int c = ((int)t & 31) * 2;
    int s = src[e];
    int d = dst[e];
    const float2 v = *(const float2*)(feat + (long long)s * D_IN + c);
    float* p = h + (long long)d * D_IN + c;
    __hip_atomic_fetch_add(p,     v.x, __ATOMIC_RELAXED, __HIP_MEMORY_SCOPE_AGENT);
    __hip_atomic_fetch_add(p + 1, v.y, __ATOMIC_RELAXED, __HIP_MEMORY_SCOPE_AGENT);
}

// ---------------------------------------------------------------------------
// Kernel C: out = h @ W.T + b  via V_WMMA_F32_16X16X4_F32 (exact f32 path).
//
// Persistent-B waves: each wave owns one 16-col N-tile, loads all 16 B
// fragments (32 VGPRs) of W for that tile ONCE, then sweeps M_TILES_PER_WAVE
// consecutive 16-row M-tiles -> 1 A-load (b64) per WMMA in the hot loop.
//
// Fragment layouts (ISA 7.12.2, 32-bit operands, wave32):
//   A 16x4 : lanes 0-15 -> M=lane,    VGPR0=K+0, VGPR1=K+1
//            lanes16-31 -> M=lane-16, VGPR0=K+2, VGPR1=K+3
//   B 4x16 : VGPR0: lanes0-15 row K+0 (N=lane), lanes16-31 row K+2
//            VGPR1: lanes0-15 row K+1,          lanes16-31 row K+3
//   C/D    : VGPR i: lanes0-15 -> (M=i, N=lane); lanes16-31 -> (M=i+8, N=lane-16)
// ---------------------------------------------------------------------------
__global__ void rrgc_gemm_wmma(const float* __restrict__ h,
                               const float* __restrict__ W,    // [D_OUT, D_IN]
                               const float* __restrict__ bias, // [D_OUT]
                               float*       __restrict__ out)  // [N_NODES, D_OUT]
{
    const int wave  = (blockIdx.x * blockDim.x + threadIdx.x) >> 5;
    const int lane  = threadIdx.x & 31;
    const int tn    = wave & 7;                     // N-tile: 0..7
    const int chunk = wave >> 3;                    // M-chunk: 0..1249
    const int mtile0 = chunk * M_TILES_PER_WAVE;
    if (mtile0 >= N_NODES / 16) return;             // uniform per wave (never taken)

    const int n0   = tn * 16;
    const int half = lane >> 4;                     // 0 or 1
    const int l15  = lane & 15;
    const int col  = n0 + l15;

    // ---- Load B fragments once: B[k][n] = W[n][k] (column n0+l15 of W.T) ----
    const float* brow = W + (long long)col * D_IN;
    v2f bfrag[D_IN / 4];
    #pragma unroll
    for (int ks = 0; ks < D_IN / 4; ++ks) {
        const int kb = ks * 4 + half * 2;
        bfrag[ks] = v2f{ brow[kb], brow[kb + 1] };
    }
    const float bv = bias[col];

    // ---- Sweep M-tiles, 1 A-load per WMMA ----
    for (int mt = 0; mt < M_TILES_PER_WAVE; ++mt) {
        const int m0 = (mtile0 + mt) * 16;
        const float* arow = h + (long long)(m0 + l15) * D_IN;

        v8f acc = {};
        #pragma unroll
        for (int ks = 0; ks < D_IN / 4; ++ks) {
            const int kb = ks * 4 + half * 2;
            v2f a = v2f{ arow[kb], arow[kb + 1] };
            acc = __builtin_amdgcn_wmma_f32_16x16x4_f32(
                      /*neg_a=*/false, a, /*neg_b=*/false, bfrag[ks],
                      /*c_mod=*/(short)0, acc, /*reuse_a=*/false, /*reuse_b=*/false);
        }

        const int r0 = m0 + half * 8;
        #pragma unroll
        for (int i = 0; i < 8; ++i) {
            out[(long long)(r0 + i) * D_OUT + col] = acc[i] + bv;
        }
    }
}

// ---------------------------------------------------------------------------
extern "C" void kernel_launch(void* const* d_in, const int* in_sizes, int n_in,
                              void* d_out, int out_size, void* d_ws, size_t ws_size,
                              hipStream_t stream) {
    const float* feat   = (const float*)d_in[0];   // [N_NODES, D_IN]
    // d_in[1] = radius: dead code in reference (weight == 1), unused.
    const int*   src    = (const int*)d_in[2];     // [N_EDGES]
    const int*   dst    = (const int*)d_in[3];     // [N_EDGES]
    const float* W      = (const float*)d_in[4];   // [D_OUT, D_IN]
    const float* bias   = (const float*)d_in[5];   // [D_OUT]
    float*       out    = (float*)d_out;           // [N_NODES, D_OUT]
    float*       h      = (float*)d_ws;            // [N_NODES, D_IN] = 25.6 MB

    // A: h = feat
    {
        const int n4 = N_NODES * D_IN / 4;
        rrgc_init_h<<<(n4 + 255) / 256, 256, 0, stream>>>(feat, h);
    }
    // B: scatter-add messages
    {
        const long long total = (long long)N_EDGES * (D_IN / 2);
        const int blocks = (int)((total + 255) / 256);
        rrgc_scatter<<<blocks, 256, 0, stream>>>(feat, src, dst, h);
    }
    // C: WMMA GEMM + bias (persistent-B waves)
    {
        const int m_chunks = (N_NODES / 16) / M_TILES_PER_WAVE;  // 1250
        const int waves    = m_chunks * (D_OUT / 16);            // 10000
        const int blocks   = (waves * 32 + 255) / 256;           // 1250
        rrgc_gemm_wmma<<<blocks, 256, 0, stream>>>(h, W, bias, out);
    }
}